// MultiHeadAttention_53188874993771
// MI455X (gfx1250) — compile-verified
//
#include <hip/hip_runtime.h>
#include <hip/hip_bf16.h>

// ---------------------------------------------------------------------------
// MultiHeadAttention with top-k sparsification, MI455X / gfx1250.
// Exact-f32 path using V_WMMA_F32_16X16X4_F32 (matches f32 reference numerics;
// the 256MB attn output write is the roofline floor anyway, ~11us @ 23.3TB/s).
// ---------------------------------------------------------------------------

typedef __attribute__((ext_vector_type(2))) float v2f;
typedef __attribute__((ext_vector_type(4))) float v4f;
typedef __attribute__((ext_vector_type(8))) float v8f;

#define WMMA_F32_4(a, b, c) \
  __builtin_amdgcn_wmma_f32_16x16x4_f32(false, (a), false, (b), (short)0, (c), false, false)

static constexpr int Bc = 2, LQc = 2048, LKc = 2048, EDc = 512;
static constexpr int Hc = 8, DKc = 64, DVc = 64, TOPKc = 64;

__device__ __forceinline__ unsigned fkey(float f) {
  // monotonic float -> uint mapping (order-preserving for radix select)
  unsigned u = __float_as_uint(f);
  return u ^ ((u & 0x80000000u) ? 0xFFFFFFFFu : 0x80000000u);
}

// ---------------------------------------------------------------------------
// Generic C[M,N] = scale * (A[M,K] @ B[K,N]); one wave per 16x16 tile.
// A layout per WMMA f32 16x16x4: lane l holds row (l&15), K = k0+2*(l>>4)+j.
// B layout: lane l holds col (l&15), K = k0+2*(l>>4)+j.
// C layout: VGPR r -> (M = r + 8*(l>>4), N = l&15).
// ---------------------------------------------------------------------------
__global__ __launch_bounds__(32) void gemm_wmma16(
    const float* __restrict__ A, const float* __restrict__ Bm, float* __restrict__ C,
    int lda, int ldb, int ldc, int K, float scale) {
  const int m0 = blockIdx.x * 16, n0 = blockIdx.y * 16;
  const int lane = threadIdx.x & 31;
  const int lo = lane & 15, lh = lane >> 4;
  const float* arow = A + (size_t)(m0 + lo) * lda;
  const float* bcol = Bm + n0 + lo;
  v8f acc = {};
  for (int k0 = 0; k0 < K; k0 += 4) {
    const int kk = k0 + 2 * lh;
    v2f a;  a.x = arow[kk];                 a.y = arow[kk + 1];
    v2f bb; bb.x = bcol[(size_t)kk * ldb];  bb.y = bcol[(size_t)(kk + 1) * ldb];
    acc = WMMA_F32_4(a, bb, acc);
  }
  for (int r = 0; r < 8; ++r)
    C[(size_t)(m0 + r + 8 * lh) * ldc + n0 + lo] = acc[r] * scale;
}

// ---------------------------------------------------------------------------
// Fused attention: scores (WMMA) -> +masks -> exact top-64 (radix select with
// index-order tie-break) -> softmax -> attn row write (NT) -> sparse attn@V.
// One 256-thread WG per (b, h, 16-query block). LDS strip: 16 x 2048 f32.
// ---------------------------------------------------------------------------
__global__ __launch_bounds__(256) void attn_fused_kernel(
    const float* __restrict__ QH, const float* __restrict__ KH, const float* __restrict__ VH,
    const float* __restrict__ qk_mask, const float* __restrict__ k_mask,
    float* __restrict__ attnOut, float* __restrict__ OH) {
  extern __shared__ float smem[];
  float*    sS     = smem;                  // 16*2048 scores/probs
  float*    sQ     = sS + 16 * 2048;        // 16*68 padded Q tile
  float*    sRed   = sQ + 16 * 68;          // 256 reduction scratch
  float*    sListP = sRed + 256;            // 128 selected probs
  unsigned* sHist  = (unsigned*)(sListP + 128);  // 256 radix bins
  unsigned* sEqOff = sHist + 256;           // 256 tie-rank exclusive scan
  unsigned* sListI = sEqOff + 256;          // 128 selected key indices
  unsigned* sB     = sListI + 128;          // 8 broadcast slots

  const int wg = blockIdx.x;
  const int qblk = wg & 127;          // LQ/16 = 128
  const int h = (wg >> 7) & 7;
  const int b = wg >> 10;
  const int q0 = qblk * 16;
  const int t = threadIdx.x;
  const int lane = t & 31, wave = t >> 5;
  const int lo = lane & 15, lh = lane >> 4;

  // Load (pre-scaled) Q tile: rows q0..q0+15, head h columns.
  for (int i = t; i < 16 * 64; i += 256) {
    const int r = i >> 6, d = i & 63;
    sQ[r * 68 + d] = QH[((size_t)(b * LQc + q0 + r)) * (Hc * DKc) + h * DKc + d];
  }
  __syncthreads();

  const float* KHb = KH + (size_t)b * LKc * (Hc * DKc) + h * DKc;
  const float* VHb = VH + (size_t)b * LKc * (Hc * DVc) + h * DVc;
  const float* qkm = qk_mask + ((size_t)b * LQc + q0) * LKc;
  const float* km  = k_mask + (size_t)b * LKc;

  // Score strip: each wave owns key tiles kt = wave, wave+8, ...
  for (int kt = wave; kt < LKc / 16; kt += 8) {
    const int kb = kt * 16;
    if (kt + 8 < LKc / 16)
      __builtin_prefetch(KHb + (size_t)((kt + 8) * 16 + lo) * (Hc * DKc), 0, 0);
    const float* qrow = sQ + lo * 68;                       // A: row = lane&15
    const float* krow = KHb + (size_t)(kb + lo) * (Hc * DKc);  // B: col = lane&15
    v8f acc = {};
    for (int k0 = 0; k0 < DKc; k0 += 4) {
      const int kk = k0 + 2 * lh;
      v2f a;  a.x = qrow[kk];  a.y = qrow[kk + 1];
      v2f bb; bb.x = krow[kk]; bb.y = krow[kk + 1];
      acc = WMMA_F32_4(a, bb, acc);
    }
    for (int r = 0; r < 8; ++r) {
      const int m = r + 8 * lh;
      const int kc = kb + lo;
      sS[m * 2048 + kc] = acc[r] + qkm[(size_t)m * LKc + kc] + km[kc];
    }
  }
  __syncthreads();

  const size_t attnBase = ((size_t)(b * Hc + h) * LQc + q0) * LKc;
  float* ohBase = OH + ((size_t)(b * Hc + h) * LQc + q0) * DVc;

  for (int r = 0; r < 16; ++r) {
    float* row = sS + r * 2048;
    const int j0 = t * 8;  // blocked ownership -> ascending index order

    // 1) row max (softmax stabilizer; max is always within the top-k set)
    float mx = -3.4e38f;
    for (int j = 0; j < 8; ++j) mx = fmaxf(mx, row[j0 + j]);
    sRed[t] = mx; __syncthreads();
    for (int s = 128; s > 0; s >>= 1) { if (t < s) sRed[t] = fmaxf(sRed[t], sRed[t + s]); __syncthreads(); }
    const float M = sRed[0];

    // 2) radix select: exact key of the 64th-largest element
    unsigned prefix = 0; int kneed = TOPKc;
    for (int byte = 3; byte >= 0; --byte) {
      sHist[t] = 0; __syncthreads();
      const unsigned hm = (byte == 3) ? 0u : (0xFFFFFFFFu << ((byte + 1) * 8));
      for (int j = 0; j < 8; ++j) {
        const unsigned key = fkey(row[j0 + j]);
        if ((key & hm) == prefix) atomicAdd(&sHist[(key >> (byte * 8)) & 255u], 1u);
      }
      __syncthreads();
      if (t == 0) {
        unsigned cum = 0; unsigned chosen = 0; int kn = kneed;
        for (int bin = 255; bin >= 0; --bin) {
          const unsigned c = sHist[bin];
          if (cum + c >= (unsigned)kneed) { chosen = (unsigned)bin; kn = kneed - (int)cum; break; }
          cum += c;
        }
        sB[0] = chosen; sB[1] = (unsigned)kn;
      }
      __syncthreads();
      prefix |= (sB[0] << (byte * 8));
      kneed = (int)sB[1];
    }
    const unsigned T = prefix;   // key of 64th largest
    const int kEq = kneed;       // #elements == T to keep (ascending index)

    // 3) tie-rank exclusive scan (matches top_k's first-index tie-break)
    unsigned eq = 0;
    for (int j = 0; j < 8; ++j) if (fkey(row[j0 + j]) == T) ++eq;
    sEqOff[t] = eq;
    if (t == 0) sB[2] = 0u;  // compact-list counter
    __syncthreads();
    if (t == 0) { unsigned runv = 0; for (int i = 0; i < 256; ++i) { unsigned c = sEqOff[i]; sEqOff[i] = runv; runv += c; } }
    __syncthreads();
    unsigned tr = sEqOff[t];

    // 4) select, exp, partial sum, compact (k,p) list
    float lsum = 0.f;
    for (int j = 0; j < 8; ++j) {
      const float v = row[j0 + j];
      const unsigned key = fkey(v);
      bool sel = key > T;
      if (key == T) { if (tr < (unsigned)kEq) sel = true; ++tr; }
      float e = 0.f;
      if (sel) {
        e = __expf(v - M);   // non-selected: exp(-10000-M) == +0.0f in f32, as reference
        const unsigned slot = atomicAdd(&sB[2], 1u);
        if (slot < 128u) { sListI[slot] = (unsigned)(j0 + j); sListP[slot] = e; }
        lsum += e;
      }
      row[j0 + j] = e;
    }
    sRed[t] = lsum; __syncthreads();
    for (int s = 128; s > 0; s >>= 1) { if (t < s) sRed[t] += sRed[t + s]; __syncthreads(); }
    const float invS = 1.0f / sRed[0];
    unsigned cnt = sB[2]; if (cnt > 128u) cnt = 128u;
    __syncthreads();

    // 5) stream full attn row to HBM (bulk of traffic) — nontemporal float4
    float* dst = attnOut + attnBase + (size_t)r * LKc;
    for (int j = 0; j < 8; j += 4) {
      v4f val;
      val.x = row[j0 + j + 0] * invS; val.y = row[j0 + j + 1] * invS;
      val.z = row[j0 + j + 2] * invS; val.w = row[j0 + j + 3] * invS;
      __builtin_nontemporal_store(val, (v4f*)(dst + j0 + j));
    }

    // 6) sparse attn @ V: 64 lanes, each one dv column; per i the 64 lanes read
    //    a contiguous 256B slice of VH (coalesced).
    if (t < DVc) {
      float acc = 0.f;
      for (unsigned i = 0; i < cnt; ++i)
        acc += sListP[i] * VHb[(size_t)sListI[i] * (Hc * DVc) + t];
      ohBase[(size_t)r * DVc + t] = acc * invS;
    }
    __syncthreads();
  }
}

// ---------------------------------------------------------------------------
// out[m,n] = sum_c OHcat[m,c] * w_fc[c,n] + v[m,n]   (m = b*LQ+q, c = h*64+d)
// ---------------------------------------------------------------------------
__global__ __launch_bounds__(32) void out_proj_kernel(
    const float* __restrict__ OH, const float* __restrict__ Wfc,
    const float* __restrict__ Vres, float* __restrict__ Out) {
  const int m0 = blockIdx.x * 16, n0 = blockIdx.y * 16;
  const int lane = threadIdx.x & 31;
  const int lo = lane & 15, lh = lane >> 4;
  const int m = m0 + lo, b = m >> 11, q = m & 2047;
  v8f acc = {};
  for (int k0 = 0; k0 < Hc * DVc; k0 += 4) {
    const int kk = k0 + 2 * lh, kk1 = kk + 1;
    v2f a;
    a.x = OH[((size_t)((b * Hc + (kk  >> 6)) * LQc + q)) * DVc + (kk  & 63)];
    a.y = OH[((size_t)((b * Hc + (kk1 >> 6)) * LQc + q)) * DVc + (kk1 & 63)];
    v2f bb; bb.x = Wfc[(size_t)kk * DVc + n0 + lo]; bb.y = Wfc[(size_t)kk1 * DVc + n0 + lo];
    acc = WMMA_F32_4(a, bb, acc);
  }
  for (int r = 0; r < 8; ++r) {
    const int mm = m0 + r + 8 * lh;
    Out[(size_t)mm * DVc + n0 + lo] = acc[r] + Vres[(size_t)mm * DVc + n0 + lo];
  }
}

// ---------------------------------------------------------------------------
extern "C" void kernel_launch(void* const* d_in, const int* in_sizes, int n_in,
                              void* d_out, int out_size, void* d_ws, size_t ws_size,
                              hipStream_t stream) {
  const float* q       = (const float*)d_in[0];
  const float* k       = (const float*)d_in[1];
  const float* v       = (const float*)d_in[2];
  // d_in[3]=qpos, d_in[4]=kpos are accepted but unused (as in reference)
  const float* qk_mask = (const float*)d_in[5];
  const float* k_mask  = (const float*)d_in[6];
  const float* w_qs    = (const float*)d_in[7];
  const float* w_ks    = (const float*)d_in[8];
  const float* w_vs    = (const float*)d_in[9];
  const float* w_fc    = (const float*)d_in[10];

  float* out  = (float*)d_out;                        // [B, LQ, DV]
  float* attn = out + (size_t)Bc * LQc * DVc;         // [B, H, LQ, LK]

  const size_t ROWS = (size_t)Bc * LQc;               // 4096
  float* QH = (float*)d_ws;                           // [4096, 512] (pre-scaled)
  float* KH = QH + ROWS * (size_t)(Hc * DKc);         // [4096, 512]
  float* VH = KH + ROWS * (size_t)(Hc * DKc);         // [4096, 512]
  float* OH = VH + ROWS * (size_t)(Hc * DVc);         // [B,H,LQ,64]

  const dim3 wblk(32);
  const float scale = 0.125f;  // 1/sqrt(DK)
  gemm_wmma16<<<dim3(ROWS / 16, (Hc * DKc) / 16), wblk, 0, stream>>>(
      q, w_qs, QH, EDc, Hc * DKc, Hc * DKc, EDc, scale);
  gemm_wmma16<<<dim3(ROWS / 16, (Hc * DKc) / 16), wblk, 0, stream>>>(
      k, w_ks, KH, EDc, Hc * DKc, Hc * DKc, EDc, 1.0f);
  gemm_wmma16<<<dim3(ROWS / 16, (Hc * DVc) / 16), wblk, 0, stream>>>(
      v, w_vs, VH, DVc, Hc * DVc, Hc * DVc, DVc, 1.0f);

  const size_t shmem = (16 * 2048 + 16 * 68 + 256 + 128) * sizeof(float) +
                       (256 + 256 + 128 + 8) * sizeof(unsigned);
  attn_fused_kernel<<<dim3(Bc * Hc * (LQc / 16)), dim3(256), shmem, stream>>>(
      QH, KH, VH, qk_mask, k_mask, attn, OH);

  out_proj_kernel<<<dim3(ROWS / 16, DVc / 16), wblk, 0, stream>>>(OH, w_fc, v, out);
}